// GraphConv_69475390980371
// MI455X (gfx1250) — compile-verified
//
#include <hip/hip_runtime.h>
#include <hip/hip_bf16.h>

typedef __attribute__((ext_vector_type(2))) float v2f;
typedef __attribute__((ext_vector_type(8))) float v8f;

#define IN_C 128
#define OUT_C 64

// ---------------------------------------------------------------------------
// Degree: deg[i] = 1 (self loop) + #edges with dst == i
// ---------------------------------------------------------------------------
__global__ void deg_init_kernel(float* __restrict__ deg, int n) {
    int i = blockIdx.x * blockDim.x + threadIdx.x;
    if (i < n) deg[i] = 1.0f;
}

__global__ void deg_acc_kernel(const long long* __restrict__ ei,
                               float* __restrict__ deg, int E) {
    int e = blockIdx.x * blockDim.x + threadIdx.x;
    if (e < E) {
        long long dst = ei[(long long)E + e];
        unsafeAtomicAdd(&deg[dst], 1.0f);
    }
}

__global__ void deg_rsqrt_kernel(float* __restrict__ deg, int n) {
    int i = blockIdx.x * blockDim.x + threadIdx.x;
    if (i < n) {
        float d = deg[i];
        deg[i] = (d > 0.0f) ? rsqrtf(d) : 0.0f;  // becomes dinv in place
    }
}

// ---------------------------------------------------------------------------
// GEMM: xw[N,64] = x[N,128] @ w[128,64] via V_WMMA_F32_16X16X4_F32.
// One wave owns a 16x64 output slab: 4 accumulators (16x16 tiles), A fragment
// shared across the 4 WMMAs per k-step. K loop: 128/4 = 32 steps.
//
// ISA layouts (32-bit):
//   A 16x4 : lanes 0-15 row M=lane  {v0:K=k0, v1:K=k0+1}
//            lanes 16-31 row M=lane-16 {v0:K=k0+2, v1:K=k0+3}
//   B 4x16 : lanes 0-15 col N=lane  {v0:K=k0, v1:K=k0+1}
//            lanes 16-31 col N=lane-16 {v0:K=k0+2, v1:K=k0+3}
//   C/D 16x16: VGPR v -> M = v + 8*(lane/16), N = lane%16
// ---------------------------------------------------------------------------
__global__ void gcn_gemm_wmma_kernel(const float* __restrict__ x,
                                     const float* __restrict__ w,
                                     float* __restrict__ xw, int nrows) {
    const int waves_per_block = blockDim.x >> 5;
    const int wave_id = blockIdx.x * waves_per_block + (threadIdx.x >> 5);
    const int m0 = wave_id * 16;
    if (m0 >= nrows) return;                 // wave-uniform: EXEC stays all-1s

    const int lane = threadIdx.x & 31;
    const int half = lane >> 4;              // 0 or 1
    const int lr   = lane & 15;

    v8f acc0 = {}, acc1 = {}, acc2 = {}, acc3 = {};

    const float* arow = x + (size_t)(m0 + lr) * IN_C;

    for (int k0 = 0; k0 < IN_C; k0 += 4) {
        const int ka = k0 + 2 * half;        // this lane's pair of K values
        v2f a;
        a.x = arow[ka];
        a.y = arow[ka + 1];

        const float* wr0 = w + (size_t)ka * OUT_C;        // row K=ka
        const float* wr1 = wr0 + OUT_C;                   // row K=ka+1
        v2f b0; b0.x = wr0[lr];      b0.y = wr1[lr];
        v2f b1; b1.x = wr0[16 + lr]; b1.y = wr1[16 + lr];
        v2f b2; b2.x = wr0[32 + lr]; b2.y = wr1[32 + lr];
        v2f b3; b3.x = wr0[48 + lr]; b3.y = wr1[48 + lr];

        acc0 = __builtin_amdgcn_wmma_f32_16x16x4_f32(false, a, false, b0,
                                                     (short)0, acc0, false, false);
        acc1 = __builtin_amdgcn_wmma_f32_16x16x4_f32(false, a, false, b1,
                                                     (short)0, acc1, false, false);
        acc2 = __builtin_amdgcn_wmma_f32_16x16x4_f32(false, a, false, b2,
                                                     (short)0, acc2, false, false);
        acc3 = __builtin_amdgcn_wmma_f32_16x16x4_f32(false, a, false, b3,
                                                     (short)0, acc3, false, false);
    }

    float* orow = xw + (size_t)m0 * OUT_C;
#pragma unroll
    for (int v = 0; v < 8; ++v) {
        const int m = v + 8 * half;
        orow[(size_t)m * OUT_C + lr]      = acc0[v];
        orow[(size_t)m * OUT_C + 16 + lr] = acc1[v];
        orow[(size_t)m * OUT_C + 32 + lr] = acc2[v];
        orow[(size_t)m * OUT_C + 48 + lr] = acc3[v];
    }
}

// ---------------------------------------------------------------------------
// out[i][c] = bias[c] + xw[i][c] * dinv[i]^2   (self-loop message, no atomics)
// ---------------------------------------------------------------------------
__global__ void init_out_kernel(const float* __restrict__ xw,
                                const float* __restrict__ dinv,
                                const float* __restrict__ bias,
                                float* __restrict__ out, int n) {
    int t = blockIdx.x * blockDim.x + threadIdx.x;
    if (t < n * OUT_C) {
        int i = t >> 6;
        int c = t & 63;
        float di = dinv[i];
        out[t] = bias[c] + xw[t] * di * di;
    }
}

// ---------------------------------------------------------------------------
// Edge scatter: one wave per edge; lane handles 2 channels.
// out[dst][c] += xw[src][c] * dinv[src] * dinv[dst]  via global_atomic_add_f32
// ---------------------------------------------------------------------------
__global__ void scatter_kernel(const long long* __restrict__ ei,
                               const float* __restrict__ xw,
                               const float* __restrict__ dinv,
                               float* __restrict__ out, int E) {
    const int wid = blockIdx.x * (blockDim.x >> 5) + (threadIdx.x >> 5);
    if (wid >= E) return;
    const int lane = threadIdx.x & 31;

    const long long j = ei[wid];                 // src
    const long long i = ei[(long long)E + wid];  // dst
    const float nrm = dinv[j] * dinv[i];

    const float* mrow = xw + (size_t)j * OUT_C;
    float* orow = out + (size_t)i * OUT_C;

    const int c = lane * 2;
    float m0 = mrow[c]     * nrm;
    float m1 = mrow[c + 1] * nrm;
    unsafeAtomicAdd(&orow[c],     m0);
    unsafeAtomicAdd(&orow[c + 1], m1);
}

// ---------------------------------------------------------------------------
extern "C" void kernel_launch(void* const* d_in, const int* in_sizes, int n_in,
                              void* d_out, int out_size, void* d_ws, size_t ws_size,
                              hipStream_t stream) {
    const float*     x    = (const float*)d_in[0];
    const long long* ei   = (const long long*)d_in[1];   // int64 [2, E]
    const float*     w    = (const float*)d_in[2];
    const float*     bias = (const float*)d_in[3];
    float* out = (float*)d_out;

    const int n = in_sizes[0] / IN_C;     // 100000
    const int E = in_sizes[1] / 2;        // 1600000

    // workspace: dinv [n floats], then xw [n*64 floats], 512B aligned
    char* ws = (char*)d_ws;
    float* dinv = (float*)ws;
    size_t dinv_bytes = ((size_t)n * sizeof(float) + 511) & ~(size_t)511;
    float* xw = (float*)(ws + dinv_bytes);

    // 1) degrees -> dinv (in place)
    deg_init_kernel<<<(n + 255) / 256, 256, 0, stream>>>(dinv, n);
    deg_acc_kernel<<<(E + 255) / 256, 256, 0, stream>>>(ei, dinv, E);
    deg_rsqrt_kernel<<<(n + 255) / 256, 256, 0, stream>>>(dinv, n);

    // 2) xw = x @ w   (WMMA; one wave per 16 rows, 8 waves per block)
    const int row_tiles = (n + 15) / 16;                 // 6250
    const int gemm_blocks = (row_tiles + 7) / 8;
    gcn_gemm_wmma_kernel<<<gemm_blocks, 256, 0, stream>>>(x, w, xw, n);

    // 3) out = bias + self-loop message
    const long long total = (long long)n * OUT_C;
    init_out_kernel<<<(int)((total + 255) / 256), 256, 0, stream>>>(xw, dinv, bias, out, n);

    // 4) edge scatter-add (one wave per edge, 8 waves per block)
    const int sc_blocks = (E + 7) / 8;
    scatter_kernel<<<sc_blocks, 256, 0, stream>>>(ei, xw, dinv, out, E);
}